// ViBERTgridGenerator_24137716203904
// MI455X (gfx1250) — compile-verified
//
#include <hip/hip_runtime.h>

typedef __attribute__((ext_vector_type(2))) float v2f;
typedef __attribute__((ext_vector_type(8))) float v8f;
typedef __attribute__((ext_vector_type(4))) float v4f;
typedef __attribute__((ext_vector_type(4))) int   v4i;

#define BS 8
#define T  512
#define D  768
#define HG 128
#define WG 96
#define NCELL (HG * WG)

// ---------------------------------------------------------------------------
// K1: per-batch preprocessing. One block per batch, 512 threads (= tokens).
//  - valid[t] = t < first_zero(mask)
//  - word segmentation via coord comparison, word_id = scan(new_word)-1
//  - meta[t] = valid ? word_id : -1 ; inv_cnt[w] = 1/max(cnt_w,1)
//  - word_start[w] = first token of word w ; binfo = {nwords, first_zero}
//  - write events (valid word starts) compacted in t-order, then each grid
//    cell scans events ascending (last cover wins) -> prevmap = prev word
//    id to paint, or -1 for "zero".
// ---------------------------------------------------------------------------
__global__ __launch_bounds__(512) void k1_prep(
    const int* __restrict__ coors, const int* __restrict__ mask,
    int* __restrict__ meta, float* __restrict__ inv_cnt,
    int* __restrict__ wstart, int* __restrict__ binfo,
    int* __restrict__ prevmap) {
  __shared__ int sc[T][4];
  __shared__ int sscan[T];
  __shared__ int scnt[T];
  __shared__ int evx0[T], evy0[T], evx1[T], evy1[T], evpw[T];
  __shared__ int sfz, snev;

  const int b = blockIdx.x;
  const int t = threadIdx.x;

  const int* cb = coors + (b * T + t) * 4;
  const int c0 = cb[0], c1 = cb[1], c2 = cb[2], c3 = cb[3];
  sc[t][0] = c0; sc[t][1] = c1; sc[t][2] = c2; sc[t][3] = c3;
  scnt[t] = 0;
  const int mk = mask[b * T + t];
  if (t == 0) sfz = T;
  __syncthreads();
  if (mk == 0) atomicMin(&sfz, t);
  __syncthreads();
  const int fz = sfz;
  const bool valid = t < fz;

  int nw;
  if (t == 0) nw = 1;
  else nw = (sc[t-1][0] != c0 || sc[t-1][1] != c1 ||
             sc[t-1][2] != c2 || sc[t-1][3] != c3) ? 1 : 0;

  // inclusive scan of new_word -> word_id
  sscan[t] = nw;
  __syncthreads();
  for (int off = 1; off < T; off <<= 1) {
    int v = (t >= off) ? sscan[t - off] : 0;
    __syncthreads();
    sscan[t] += v;
    __syncthreads();
  }
  const int wid = sscan[t] - 1;
  if (t == T - 1) binfo[b * 2 + 0] = sscan[t];  // nwords
  if (t == 0)     binfo[b * 2 + 1] = fz;

  meta[b * T + t] = valid ? wid : -1;
  if (nw) wstart[b * T + wid] = t;
  if (valid) atomicAdd(&scnt[wid], 1);
  __syncthreads();
  {
    int cw = scnt[t];
    inv_cnt[b * T + t] = 1.0f / (float)(cw < 1 ? 1 : cw);
  }

  // compact write events (valid word starts) in token order
  const int iw = (nw && valid) ? 1 : 0;
  sscan[t] = iw;
  __syncthreads();
  for (int off = 1; off < T; off <<= 1) {
    int v = (t >= off) ? sscan[t - off] : 0;
    __syncthreads();
    sscan[t] += v;
    __syncthreads();
  }
  if (t == T - 1) snev = sscan[t];
  if (iw) {
    const int pos = sscan[t] - 1;
    evx0[pos] = c0 >> 3; evy0[pos] = c1 >> 3;  // STRIDE = 8
    evx1[pos] = c2 >> 3; evy1[pos] = c3 >> 3;
    evpw[pos] = wid - 1;                       // value painted = prev word
  }
  __syncthreads();

  const int nev = snev;
  for (int cell = t; cell < NCELL; cell += T) {
    const int y = cell / WG, x = cell % WG;
    int pw = -1;
    for (int e = 0; e < nev; ++e) {
      if (x >= evx0[e] && x < evx1[e] && y >= evy0[e] && y < evy1[e])
        pw = evpw[e];  // ascending t -> last cover wins (argmax t)
    }
    prevmap[b * NCELL + cell] = pw;
  }
}

// ---------------------------------------------------------------------------
// K2: word means via V_WMMA_F32_16X16X4_F32 (exact fp32 accumulate).
//  One wave per (16-word x 16-dim) tile. A = one-hot word-membership
//  (valid gated through meta == -1), B = embedding slab, K looped in steps
//  of 4 tokens restricted to the word tile's contiguous token span.
//  A layout: m = lane&15, VGPR0/1 = K = (lane>>4)*2, +1.
//  B layout: n = lane&15, VGPR0/1 = K = (lane>>4)*2, +1.
//  C layout: VGPR i -> M = i + (lane>>4)*8, N = lane&15.
// ---------------------------------------------------------------------------
__global__ __launch_bounds__(32) void k2_wordmean(
    const float* __restrict__ emb, const int* __restrict__ meta,
    const float* __restrict__ inv_cnt, const int* __restrict__ wstart,
    const int* __restrict__ binfo, float* __restrict__ wm) {
  const int dtile = blockIdx.x;       // 0..47
  const int wtile = blockIdx.y;       // 0..31
  const int b     = blockIdx.z;       // 0..7
  const int lane  = threadIdx.x;      // 0..31
  const int mrow  = lane & 15;
  const int hi    = lane >> 4;        // 0 or 1
  const int w0    = wtile * 16;
  const int d     = dtile * 16 + mrow;
  const int nwords = binfo[b * 2 + 0];

  v8f c = {};
  if (w0 < nwords) {
    const int t0 = wstart[b * T + w0] & ~3;
    const int t1 = (w0 + 16 < nwords) ? wstart[b * T + w0 + 16] : T;
    const int* mt = meta + b * T;
    const float* eb = emb + (long)b * T * D;
    const int wme = w0 + mrow;
    for (int kb = t0; kb < t1; kb += 4) {
      const int ka = kb + hi * 2;
      v2f a, bb;
      a.x = (mt[ka]     == wme) ? 1.0f : 0.0f;
      a.y = (mt[ka + 1] == wme) ? 1.0f : 0.0f;
      bb.x = eb[(long)ka * D + d];
      bb.y = eb[(long)(ka + 1) * D + d];
      c = __builtin_amdgcn_wmma_f32_16x16x4_f32(
          /*neg_a=*/false, a, /*neg_b=*/false, bb,
          /*c_mod=*/(short)0, c, /*reuse_a=*/false, /*reuse_b=*/false);
    }
  }
  float* out = wm + (long)b * T * D;
#pragma unroll
  for (int i = 0; i < 8; ++i) {
    const int w = w0 + i + hi * 8;
    out[(long)w * D + d] = c[i] * inv_cnt[b * T + w];
  }
}

// ---------------------------------------------------------------------------
// K3: coalesced output fill (the 302 MB store that bounds this problem).
//  grid[b,d,y,x] = prevmap[b,y,x] >= 0 ? word_mean[b, pw, d] : 0
//  4 consecutive x-cells per thread -> b128 prevmap load + one b128
//  non-temporal store (output is write-once and bigger than L2; keep L2
//  for the word_mean gather table instead).
// ---------------------------------------------------------------------------
__global__ __launch_bounds__(256) void k3_fill(
    const float* __restrict__ wm, const int* __restrict__ prevmap,
    float* __restrict__ out) {
  const int q = blockIdx.x * 256 + threadIdx.x;     // quad index
  const int idx = q * 4;                            // element index
  const int cell = idx % NCELL;                     // multiple of 4
  const int rest = idx / NCELL;
  const int d = rest % D;
  const int b = rest / D;

  const v4i pw4 = *(const v4i*)(prevmap + b * NCELL + cell);
  const float* wb = wm + (long)b * T * D + d;
  v4f v;
  v.x = (pw4.x >= 0) ? wb[(long)pw4.x * D] : 0.0f;
  v.y = (pw4.y >= 0) ? wb[(long)pw4.y * D] : 0.0f;
  v.z = (pw4.z >= 0) ? wb[(long)pw4.z * D] : 0.0f;
  v.w = (pw4.w >= 0) ? wb[(long)pw4.w * D] : 0.0f;
  __builtin_nontemporal_store(v, (v4f*)(out + idx));
}

extern "C" void kernel_launch(void* const* d_in, const int* in_sizes, int n_in,
                              void* d_out, int out_size, void* d_ws,
                              size_t ws_size, hipStream_t stream) {
  const float* emb  = (const float*)d_in[0];  // [8,512,768] f32
  const int* coors  = (const int*)d_in[1];    // [8,512,4]
  const int* mask   = (const int*)d_in[2];    // [8,512]
  float* out = (float*)d_out;                 // [8,768,128,96] f32

  char* ws = (char*)d_ws;
  size_t o = 0;
  float* wm      = (float*)(ws + o); o += (size_t)BS * T * D * 4;   // 12.6 MB
  int*   prevmap = (int*)  (ws + o); o += (size_t)BS * NCELL * 4;
  int*   meta    = (int*)  (ws + o); o += (size_t)BS * T * 4;
  float* invc    = (float*)(ws + o); o += (size_t)BS * T * 4;
  int*   wst     = (int*)  (ws + o); o += (size_t)BS * T * 4;
  int*   binfo   = (int*)  (ws + o); o += (size_t)BS * 2 * 4;

  k1_prep<<<BS, T, 0, stream>>>(coors, mask, meta, invc, wst, binfo, prevmap);
  k2_wordmean<<<dim3(D / 16, T / 16, BS), 32, 0, stream>>>(emb, meta, invc,
                                                           wst, binfo, wm);
  const int nquads = BS * D * NCELL / 4;  // 18,874,368
  k3_fill<<<nquads / 256, 256, 0, stream>>>(wm, prevmap, out);
}